// AugNorm_78563541778825
// MI455X (gfx1250) — compile-verified
//
#include <hip/hip_runtime.h>
#include <cstdint>

typedef __attribute__((ext_vector_type(2))) float v2f;
typedef __attribute__((ext_vector_type(8))) float v8f;

#define EPS_C     1e-12f
#define VAR_EPS_C 1e-16f

// Column-sum-and-broadcast: D = ones(16x4) * B(4x16) + C.
// Every lane's element 0 of the accumulator ends up holding the full column
// sum for column (lane & 15) of this wave's 16-column tile.
__device__ __forceinline__ v8f wmma_colsum(v2f b, v8f c) {
  v2f a = {1.0f, 1.0f};
  return __builtin_amdgcn_wmma_f32_16x16x4_f32(
      /*neg_a=*/false, a, /*neg_b=*/false, b,
      /*c_mod=*/(short)0, c, /*reuse_a=*/false, /*reuse_b=*/false);
}

__global__ __launch_bounds__(256) void augnorm_kernel(
    const float* __restrict__ X, const float* __restrict__ W,
    const float* __restrict__ Bv, float* __restrict__ out)
{
  extern __shared__ float tile[];   // 128 x 128 f32 = 64 KB, same layout as global

  const int t    = threadIdx.x;
  const int lane = t & 31;
  const int wv   = t >> 5;          // wave id 0..7 -> columns [16*wv, 16*wv+16)
  const int l15  = lane & 15;
  const int hi   = (lane >> 4) & 1; // half-wave selects h parity (WMMA B layout)
  const int wcol = wv * 16 + l15;

  const int    ch    = blockIdx.x;                         // channel 0..767
  const size_t plane = ((size_t)blockIdx.y * 768 + ch) * 16384u;
  const float* src   = X + plane;

  // ---- Stage whole 64KB tile: CDNA5 async global->LDS (ASYNCcnt path) ----
  {
    uint32_t lds_base = (uint32_t)(uintptr_t)tile;     // low 32 bits = LDS offset
    uint64_t gbase    = (uint64_t)(uintptr_t)src;
#pragma unroll
    for (int i = 0; i < 16; ++i) {
      uint32_t chunk = (uint32_t)t + (uint32_t)i * 256u;   // 4096 x 16B chunks
      uint32_t la = lds_base + chunk * 16u;
      uint64_t ga = gbase + (uint64_t)chunk * 16u;
      asm volatile("global_load_async_to_lds_b128 %0, %1, off"
                   :: "v"(la), "v"(ga) : "memory");
    }
    asm volatile("s_wait_asynccnt 0" ::: "memory");
  }
  __syncthreads();

  // ---- Pull this wave's 16 columns into registers, pre-shaped as WMMA B ----
  // reg j: lanes 0-15 -> (h=2j, w=wcol), lanes 16-31 -> (h=2j+1, w=wcol)
  float x[64];
#pragma unroll
  for (int j = 0; j < 64; ++j)
    x[j] = tile[(2 * j + hi) * 128 + wcol];

  const v8f zero8 = {0.f, 0.f, 0.f, 0.f, 0.f, 0.f, 0.f, 0.f};

  // ---- mean over h via WMMA ones-reduction (broadcast to all lanes) ----
  v8f acc = zero8;
#pragma unroll
  for (int k = 0; k < 32; ++k) {
    v2f bvec = { x[2 * k], x[2 * k + 1] };
    acc = wmma_colsum(bvec, acc);
  }
  float y = acc[0] * (1.0f / 128.0f);

  // ---- 4 Newton iterations for the phi=1.5 generalized median ----
  for (int it = 0; it < 4; ++it) {
    v8f aFx = zero8, aFxx = zero8;
#pragma unroll
    for (int k = 0; k < 32; ++k) {
      float d0 = y - x[2 * k];
      float d1 = y - x[2 * k + 1];
      d0 += (d0 < 0.f) ? -EPS_C : EPS_C;     // jnp.where(dev<0, dev-eps, dev+eps)
      d1 += (d1 < 0.f) ? -EPS_C : EPS_C;
      float r0 = __builtin_amdgcn_rsqf(__builtin_fabsf(d0));  // |dev|^(phi-2)
      float r1 = __builtin_amdgcn_rsqf(__builtin_fabsf(d1));
      v2f bfx  = { r0 * d0, r1 * d1 };       // F_x terms
      v2f bfxx = { r0, r1 };                 // F_xx terms
      aFx  = wmma_colsum(bfx,  aFx);
      aFxx = wmma_colsum(bfxx, aFxx);
    }
    y -= aFx[0] / (0.5f * aFxx[0]);          // F_xx = (phi-1) * sum
  }

  // ---- variance over h ----
  v8f av = zero8;
#pragma unroll
  for (int k = 0; k < 32; ++k) {
    float d0 = x[2 * k]     - y;
    float d1 = x[2 * k + 1] - y;
    v2f bsq = { d0 * d0, d1 * d1 };
    av = wmma_colsum(bsq, av);
  }
  float var = av[0] * (1.0f / 128.0f);
  float inv = __builtin_amdgcn_rsqf(var + VAR_EPS_C);

  const float wgt = W[ch];
  const float bs  = Bv[ch];

  // ---- normalize + affine, write straight back to HBM ----
  float* dst = out + plane;
#pragma unroll
  for (int j = 0; j < 64; ++j) {
    int idx = (2 * j + hi) * 128 + wcol;
    dst[idx] = fmaf((x[j] - y) * inv, wgt, bs);
  }
}

extern "C" void kernel_launch(void* const* d_in, const int* in_sizes, int n_in,
                              void* d_out, int out_size, void* d_ws, size_t ws_size,
                              hipStream_t stream) {
  const float* X = (const float*)d_in[0];   // (4,768,128,128) f32
  const float* W = (const float*)d_in[1];   // (768,) f32
  const float* B = (const float*)d_in[2];   // (768,) f32
  float* out = (float*)d_out;               // same shape as X, f32

  dim3 grid(768, 4);                        // one block per (b, c) plane
  augnorm_kernel<<<grid, dim3(256), 65536, stream>>>(X, W, B, out);
}